// GAT_40209483825154
// MI455X (gfx1250) — compile-verified
//
#include <hip/hip_runtime.h>
#include <hip/hip_bf16.h>

// ---------------- problem constants (match reference) ----------------
#define N_NODES    20000
#define N_EDGES_IN 320000
#define E_TOT      (N_EDGES_IN + N_NODES)   // edges + self loops = 340000
#define HEADS      4
#define HID        64
#define FDIM       (HEADS * HID)            // 256
#define NEG_SLOPE  0.2f

typedef __attribute__((ext_vector_type(2))) float v2f;
typedef __attribute__((ext_vector_type(8))) float v8f;

// order-preserving float<->int encoding for atomic float max
__device__ __forceinline__ int enc_f32(float f) {
  int i = __float_as_int(f);
  return (i >= 0) ? i : (i ^ 0x7fffffff);
}
__device__ __forceinline__ float dec_f32(int i) {
  return __int_as_float((i >= 0) ? i : (i ^ 0x7fffffff));
}

// src/dst with implicit self loops appended after the real edges
__device__ __forceinline__ void edge_ends(const long long* __restrict__ ei, int e,
                                          int& s, int& d) {
  if (e < N_EDGES_IN) {
    s = (int)ei[e];
    d = (int)ei[N_EDGES_IN + e];
  } else {
    s = d = e - N_EDGES_IN;
  }
}

// ======================= GEMM via V_WMMA_F32_16X16X4_F32 =======================
// Block: 64 threads (2 waves). Each wave owns a 16-row strip and 4 column tiles
// (16x64 of output) -> A fragment loaded once per k-step, reused by 4 WMMAs.
// W slab (64k x 64n) staged in LDS by the whole block, consumed via ds_load.
//
// A layout (16x4 f32): lane = half*16 + m ; vgpr0/1 = A[m][half*2 + 0/1]
// B layout (4x16 f32): lane = half*16 + n ; vgpr0/1 = B[half*2 + 0/1][n]
// C/D layout (16x16 f32): vgpr r, lane = half*16 + n -> D[r + half*8][n]
#define LDS_STRIDE 68   // %64==4: half-wave ka vs ka+2 reads land 8 banks apart; float4-aligned

__global__ void gat_gemm_wmma(const float* __restrict__ X, const float* __restrict__ W,
                              float* __restrict__ H, int K, int M) {
  __shared__ float lds[64 * LDS_STRIDE];

  const int lane = threadIdx.x & 31;
  const int wave = threadIdx.x >> 5;
  const int half = lane >> 4;
  const int l15  = lane & 15;
  const int rowBase = blockIdx.x * 32 + wave * 16;   // exact: N % 32 == 0
  const int colBase = blockIdx.y * 64;               // exact: M % 64 == 0

  const float* xrow = X + (size_t)(rowBase + l15) * (size_t)K;
  v8f acc[4] = {};

  for (int ks = 0; ks < K; ks += 64) {
    // ---- stage 64x64 W slab into LDS (coalesced float4) ----
    for (int i = threadIdx.x; i < 1024; i += 64) {   // 1024 float4 units
      const int e4  = i << 2;
      const int row = e4 >> 6;
      const int col = e4 & 63;
      const float4 w4 = *(const float4*)(W + (size_t)(ks + row) * (size_t)M + colBase + col);
      *(float4*)(&lds[row * LDS_STRIDE + col]) = w4;
    }
    __syncthreads();

    __builtin_prefetch(xrow + ks + 64, 0, 1);        // next A slab -> global_prefetch_b8

    for (int k0 = 0; k0 < 64; k0 += 4) {
      const int ka = k0 + half * 2;
      const float2 av = *(const float2*)(xrow + ks + ka);   // 8B aligned (ka even)
      v2f a; a[0] = av.x; a[1] = av.y;
#pragma unroll
      for (int t = 0; t < 4; ++t) {
        v2f b;
        b[0] = lds[(ka + 0) * LDS_STRIDE + t * 16 + l15];
        b[1] = lds[(ka + 1) * LDS_STRIDE + t * 16 + l15];
        acc[t] = __builtin_amdgcn_wmma_f32_16x16x4_f32(false, a, false, b,
                                                       (short)0, acc[t], false, false);
      }
    }
    __syncthreads();
  }

#pragma unroll
  for (int t = 0; t < 4; ++t) {
#pragma unroll
    for (int r = 0; r < 8; ++r) {
      const int row = rowBase + r + half * 8;
      H[(size_t)row * (size_t)M + colBase + t * 16 + l15] = acc[t][r];
    }
  }
}

// -------- alpha_s/alpha_d: per (node, head) dot over channels (float4) --------
__global__ void gat_alpha(const float* __restrict__ H, const float* __restrict__ a_src,
                          const float* __restrict__ a_dst, float* __restrict__ as,
                          float* __restrict__ ad) {
  const int tid = blockIdx.x * blockDim.x + threadIdx.x;
  if (tid >= N_NODES * HEADS) return;
  const int n = tid / HEADS;
  const int h = tid % HEADS;
  const float4* hp  = (const float4*)(H + (size_t)n * FDIM + (size_t)h * HID);
  const float4* asr = (const float4*)(a_src + (size_t)h * HID);
  const float4* adr = (const float4*)(a_dst + (size_t)h * HID);
  float s = 0.f, d = 0.f;
#pragma unroll 4
  for (int c = 0; c < HID / 4; ++c) {
    const float4 v = hp[c];
    const float4 ws = asr[c];
    const float4 wd = adr[c];
    s += v.x * ws.x + v.y * ws.y + v.z * ws.z + v.w * ws.w;
    d += v.x * wd.x + v.y * wd.y + v.z * wd.z + v.w * wd.w;
  }
  as[tid] = s;
  ad[tid] = d;
}

__global__ void gat_init_i32(int* __restrict__ p, long long n, int v) {
  const long long t = (long long)blockIdx.x * blockDim.x + threadIdx.x;
  if (t < n) p[t] = v;
}

// -------- edge pass 1: e = leakyrelu(as[src]+ad[dst]); seg-max via atomicMax --------
__global__ void gat_edge_max(const float* __restrict__ as, const float* __restrict__ ad,
                             const long long* __restrict__ ei, float* __restrict__ ebuf,
                             int* __restrict__ emax, int heads) {
  const long long tid = (long long)blockIdx.x * blockDim.x + threadIdx.x;
  if (tid >= (long long)E_TOT * heads) return;
  const int e = (int)(tid / heads);
  const int h = (int)(tid % heads);
  int s, d; edge_ends(ei, e, s, d);
  float v = as[s * heads + h] + ad[d * heads + h];
  v = (v > 0.f) ? v : NEG_SLOPE * v;
  ebuf[tid] = v;
  atomicMax(&emax[d * heads + h], enc_f32(v));
}

// -------- edge pass 2: ex = exp(e - max[dst]); seg-sum via atomicAdd --------
__global__ void gat_edge_expsum(float* __restrict__ ebuf, const int* __restrict__ emax,
                                float* __restrict__ denom, const long long* __restrict__ ei,
                                int heads) {
  const long long tid = (long long)blockIdx.x * blockDim.x + threadIdx.x;
  if (tid >= (long long)E_TOT * heads) return;
  const int e = (int)(tid / heads);
  const int h = (int)(tid % heads);
  int s, d; edge_ends(ei, e, s, d);
  const float ex = __expf(ebuf[tid] - dec_f32(emax[d * heads + h]));
  ebuf[tid] = ex;
  atomicAdd(&denom[d * heads + h], ex);
}

// -------- edge pass 3 (layers 1/2): vectorized float4 gather + f32 atomics --------
__global__ void gat_edge_scatter_v4(const float* __restrict__ Hfeat, const float* __restrict__ ebuf,
                                    const float* __restrict__ denom, const long long* __restrict__ ei,
                                    float* __restrict__ acc) {
  const long long tid = (long long)blockIdx.x * blockDim.x + threadIdx.x;
  if (tid >= (long long)E_TOT * (FDIM / 4)) return;
  const int e  = (int)(tid >> 6);      // FDIM/4 == 64 float4 units per edge
  const int q  = (int)(tid & 63);
  const int hc = q << 2;
  const int h  = hc >> 6;              // / HID
  int s, d; edge_ends(ei, e, s, d);
  const float alpha = ebuf[e * HEADS + h] / denom[d * HEADS + h];
  const float4 v = *(const float4*)(Hfeat + (size_t)s * FDIM + hc);
  float* ap = acc + (size_t)d * FDIM + hc;
  atomicAdd(ap + 0, v.x * alpha);
  atomicAdd(ap + 1, v.y * alpha);
  atomicAdd(ap + 2, v.z * alpha);
  atomicAdd(ap + 3, v.w * alpha);
}

// -------- edge pass 3 (layer 3, heads=1, chan=1) --------
__global__ void gat_edge_scatter_s(const float* __restrict__ Hfeat, const float* __restrict__ ebuf,
                                   const float* __restrict__ denom, const long long* __restrict__ ei,
                                   float* __restrict__ acc) {
  const int e = blockIdx.x * blockDim.x + threadIdx.x;
  if (e >= E_TOT) return;
  int s, d; edge_ends(ei, e, s, d);
  const float alpha = ebuf[e] / denom[d];
  atomicAdd(&acc[d], Hfeat[s] * alpha);
}

// -------- bias add (+optional ELU), in place capable --------
__global__ void gat_bias_act(const float* __restrict__ acc, const float* __restrict__ b,
                             float* __restrict__ out, long long total, int M, int do_elu) {
  const long long t = (long long)blockIdx.x * blockDim.x + threadIdx.x;
  if (t >= total) return;
  float v = acc[t] + b[t % M];
  if (do_elu) v = (v > 0.f) ? v : (__expf(v) - 1.0f);
  out[t] = v;
}

// -------- layer 3 matvec: h3[n] = X[n,:]·w ; alpha_s/d scalars --------
__global__ void gat_matvec3(const float* __restrict__ X, const float* __restrict__ w,
                            const float* __restrict__ asr, const float* __restrict__ adr,
                            float* __restrict__ h3, float* __restrict__ as,
                            float* __restrict__ ad, int K) {
  const int wave = threadIdx.x >> 5;
  const int lane = threadIdx.x & 31;
  const int n = blockIdx.x * (blockDim.x >> 5) + wave;
  if (n >= N_NODES) return;
  float s = 0.f;
  for (int k = lane; k < K; k += 32) s += X[(size_t)n * K + k] * w[k];
#pragma unroll
  for (int off = 16; off > 0; off >>= 1) s += __shfl_xor(s, off, 32);
  if (lane == 0) {
    h3[n] = s;
    as[n] = s * asr[0];
    ad[n] = s * adr[0];
  }
}

// ---------------------------------------------------------------------------
static inline int cdiv_ll(long long a, int b) { return (int)((a + b - 1) / b); }

extern "C" void kernel_launch(void* const* d_in, const int* in_sizes, int n_in,
                              void* d_out, int out_size, void* d_ws, size_t ws_size,
                              hipStream_t stream) {
  const float*     x  = (const float*)d_in[0];
  const long long* ei = (const long long*)d_in[1];
  const float* W1 = (const float*)d_in[2];
  const float* aS1 = (const float*)d_in[3];
  const float* aD1 = (const float*)d_in[4];
  const float* b1 = (const float*)d_in[5];
  const float* W2 = (const float*)d_in[6];
  const float* aS2 = (const float*)d_in[7];
  const float* aD2 = (const float*)d_in[8];
  const float* b2 = (const float*)d_in[9];
  const float* W3 = (const float*)d_in[10];
  const float* aS3 = (const float*)d_in[11];
  const float* aD3 = (const float*)d_in[12];
  const float* b3 = (const float*)d_in[13];
  float* out = (float*)d_out;

  // workspace carve-up
  char* ws = (char*)d_ws;
  size_t p = 0;
  auto carve = [&](size_t bytes) -> void* {
    void* q = ws + p;
    p += (bytes + 255) & ~(size_t)255;
    return q;
  };
  float* bufH  = (float*)carve((size_t)N_NODES * FDIM * 4);  // transformed features
  float* bufA  = (float*)carve((size_t)N_NODES * FDIM * 4);  // layer-1 output
  float* bufB  = (float*)carve((size_t)N_NODES * FDIM * 4);  // layer-2 output
  float* ebuf  = (float*)carve((size_t)E_TOT * HEADS * 4);   // e then exp(e-max)
  float* as    = (float*)carve((size_t)N_NODES * HEADS * 4);
  float* ad    = (float*)carve((size_t)N_NODES * HEADS * 4);
  int*   emax  = (int*)  carve((size_t)N_NODES * HEADS * 4);
  float* denom = (float*)carve((size_t)N_NODES * HEADS * 4);
  float* h3    = (float*)carve((size_t)N_NODES * 4);

  const int B = 256;
  const dim3 gemmGrid(N_NODES / 32, FDIM / 64);   // (625, 4), 64-thread blocks

  // ---------------- layer 1: in=128 -> [N,256], elu ----------------
  gat_gemm_wmma<<<gemmGrid, 64, 0, stream>>>(x, W1, bufH, 128, FDIM);
  gat_alpha<<<cdiv_ll((long long)N_NODES * HEADS, B), B, 0, stream>>>(bufH, aS1, aD1, as, ad);
  gat_init_i32<<<cdiv_ll((long long)N_NODES * HEADS, B), B, 0, stream>>>(emax, (long long)N_NODES * HEADS, (int)0x80000000);
  hipMemsetAsync(denom, 0, (size_t)N_NODES * HEADS * 4, stream);
  hipMemsetAsync(bufA, 0, (size_t)N_NODES * FDIM * 4, stream);
  gat_edge_max<<<cdiv_ll((long long)E_TOT * HEADS, B), B, 0, stream>>>(as, ad, ei, ebuf, emax, HEADS);
  gat_edge_expsum<<<cdiv_ll((long long)E_TOT * HEADS, B), B, 0, stream>>>(ebuf, emax, denom, ei, HEADS);
  gat_edge_scatter_v4<<<cdiv_ll((long long)E_TOT * (FDIM / 4), B), B, 0, stream>>>(bufH, ebuf, denom, ei, bufA);
  gat_bias_act<<<cdiv_ll((long long)N_NODES * FDIM, B), B, 0, stream>>>(bufA, b1, bufA, (long long)N_NODES * FDIM, FDIM, 1);

  // ---------------- layer 2: in=256 -> [N,256], elu ----------------
  gat_gemm_wmma<<<gemmGrid, 64, 0, stream>>>(bufA, W2, bufH, FDIM, FDIM);
  gat_alpha<<<cdiv_ll((long long)N_NODES * HEADS, B), B, 0, stream>>>(bufH, aS2, aD2, as, ad);
  gat_init_i32<<<cdiv_ll((long long)N_NODES * HEADS, B), B, 0, stream>>>(emax, (long long)N_NODES * HEADS, (int)0x80000000);
  hipMemsetAsync(denom, 0, (size_t)N_NODES * HEADS * 4, stream);
  hipMemsetAsync(bufB, 0, (size_t)N_NODES * FDIM * 4, stream);
  gat_edge_max<<<cdiv_ll((long long)E_TOT * HEADS, B), B, 0, stream>>>(as, ad, ei, ebuf, emax, HEADS);
  gat_edge_expsum<<<cdiv_ll((long long)E_TOT * HEADS, B), B, 0, stream>>>(ebuf, emax, denom, ei, HEADS);
  gat_edge_scatter_v4<<<cdiv_ll((long long)E_TOT * (FDIM / 4), B), B, 0, stream>>>(bufH, ebuf, denom, ei, bufB);
  gat_bias_act<<<cdiv_ll((long long)N_NODES * FDIM, B), B, 0, stream>>>(bufB, b2, bufB, (long long)N_NODES * FDIM, FDIM, 1);

  // ---------------- layer 3: in=256 -> [N,1], heads=1, mean == identity ----------------
  gat_matvec3<<<cdiv_ll((long long)N_NODES, 8), 256, 0, stream>>>(bufB, W3, aS3, aD3, h3, as, ad, FDIM);
  gat_init_i32<<<cdiv_ll((long long)N_NODES, B), B, 0, stream>>>(emax, (long long)N_NODES, (int)0x80000000);
  hipMemsetAsync(denom, 0, (size_t)N_NODES * 4, stream);
  hipMemsetAsync(out, 0, (size_t)N_NODES * 4, stream);
  gat_edge_max<<<cdiv_ll((long long)E_TOT, B), B, 0, stream>>>(as, ad, ei, ebuf, emax, 1);
  gat_edge_expsum<<<cdiv_ll((long long)E_TOT, B), B, 0, stream>>>(ebuf, emax, denom, ei, 1);
  gat_edge_scatter_s<<<cdiv_ll((long long)E_TOT, B), B, 0, stream>>>(h3, ebuf, denom, ei, out);
  gat_bias_act<<<cdiv_ll((long long)N_NODES, B), B, 0, stream>>>(out, b3, out, (long long)N_NODES, 1, 0);
}